// LCC_54589034332611
// MI455X (gfx1250) — compile-verified
//
#include <hip/hip_runtime.h>
#include <stdint.h>

// Problem geometry: I,J are [B=2][1][D=192][H=192][W=192] float32.
#define B_  2
#define D_  192
#define H_  192
#define W_  192
#define PL  (H_ * W_)                      // 36864, plane stride in elems
#define NVOX ((size_t)B_ * D_ * H_ * W_)   // 14,155,776

#define TH   4                 // output rows (H) per block in kernel 1
#define RIN  (TH + 8)          // 12 input rows staged (4-halo each side)
#define HT   (H_ / TH)         // 48 H-tiles
#define K1_BLOCKS (B_ * D_ * HT)     // 18432
#define K1_THREADS 192

#define SEG   48               // D-segment per thread in kernel 2
#define NSEG  (D_ / SEG)       // 4
#define K2_THREADS 256
#define K2_BLOCKS ((B_ * NSEG * H_ * W_) / K2_THREADS)   // 1152
#define PFD   8                // prefetch distance in planes for kernel 2

// ---------------- CDNA5 async global->LDS helpers ----------------
// gfx1250 async copy (tracked by ASYNCcnt). Syntax per CDNA5 ISA:
//   global_load_async_to_lds_b128 vdst(lds byte addr), vaddr(64b global), off
__device__ __forceinline__ void async_ld_b128(uint32_t lds_byte_addr, const void* gaddr) {
    asm volatile("global_load_async_to_lds_b128 %0, %1, off"
                 :: "v"(lds_byte_addr), "v"(gaddr)
                 : "memory");
}
__device__ __forceinline__ void wait_async0() {
    asm volatile("s_wait_asynccnt 0" ::: "memory");
}
// Low 32 bits of a generic pointer into LDS = wave-relative LDS byte address
// (flat LDS aperture truncates to addr[31:0], per CDNA5 ISA).
__device__ __forceinline__ uint32_t lds_addr_of(const void* p) {
    return (uint32_t)(uintptr_t)p;
}

// =================================================================
// Kernel 1: fused  (I,J) -> 5 channels, box-summed along W and H.
// One block: fixed (b,d), TH output H-rows, full W line.
// =================================================================
__global__ __launch_bounds__(K1_THREADS) void lcc_wh_kernel(
        const float* __restrict__ I, const float* __restrict__ J,
        float* __restrict__ ws)
{
    __shared__ float raw[2][RIN][W_];    // staged I,J rows (zero-padded halos)
    __shared__ float chan[5][RIN][W_];   // W-box-sums of I,J,I2,J2,IJ

    const int tid = threadIdx.x;
    const int bid = blockIdx.x;                 // ((b*D_ + d)*HT + ht)
    const int ht  = bid % HT;
    const int d   = (bid / HT) % D_;
    const int b   = bid / (HT * D_);
    const int h0  = ht * TH;

    // ---- Stage A: async-stage 12 rows x 192 floats of I and J into LDS ----
    // 2 arrays * 12 rows * 48 b128-chunks = 1152 chunks; 6 per thread.
    {
        const float* srcs[2] = { I, J };
        for (int c = tid; c < 2 * RIN * (W_ / 4); c += K1_THREADS) {
            const int a   = c / (RIN * (W_ / 4));
            const int rem = c % (RIN * (W_ / 4));
            const int r   = rem / (W_ / 4);
            const int g   = rem % (W_ / 4);       // 16-byte chunk in row
            const int h   = h0 - 4 + r;
            float* lp = &raw[a][r][g * 4];
            if (h >= 0 && h < H_) {
                const float* gp = srcs[a] +
                    ((((size_t)b * D_ + d) * H_ + h) * W_ + (size_t)g * 4);
                async_ld_b128(lds_addr_of(lp), gp);
            } else {
                // SAME padding with zero init value -> zero-fill halo rows
                float4 z = make_float4(0.f, 0.f, 0.f, 0.f);
                *(float4*)lp = z;
            }
        }
    }
    wait_async0();
    __syncthreads();

    // ---- Stage B: W-direction running box sums of the 5 channels ----
    // tid -> (r = tid/16 in 0..11, 12-wide W segment = tid%16); this mapping
    // keeps LDS bank conflicts to 2-way within a wave32.
    {
        const int r  = tid >> 4;          // 0..11
        const int w0 = (tid & 15) * 12;   // 0,12,...,180
        float si = 0.f, sj = 0.f, sii = 0.f, sjj = 0.f, sij = 0.f;
        #pragma unroll
        for (int k = -4; k <= 4; ++k) {
            const int kk = w0 + k;
            if (kk >= 0 && kk < W_) {
                const float iv = raw[0][r][kk], jv = raw[1][r][kk];
                si += iv; sj += jv; sii += iv * iv; sjj += jv * jv; sij += iv * jv;
            }
        }
        for (int w = w0; ; ) {
            chan[0][r][w] = si;  chan[1][r][w] = sj;
            chan[2][r][w] = sii; chan[3][r][w] = sjj; chan[4][r][w] = sij;
            ++w;
            if (w >= w0 + 12) break;
            const int ka = w + 4, kr = w - 5;
            if (ka < W_) {
                const float iv = raw[0][r][ka], jv = raw[1][r][ka];
                si += iv; sj += jv; sii += iv * iv; sjj += jv * jv; sij += iv * jv;
            }
            if (kr >= 0) {
                const float iv = raw[0][r][kr], jv = raw[1][r][kr];
                si -= iv; sj -= jv; sii -= iv * iv; sjj -= jv * jv; sij -= iv * jv;
            }
        }
    }
    __syncthreads();

    // ---- Stage C: H-direction 9-row running sums, store 5 volumes ----
    // tid == w column. Halo rows are already zero, so no H-edge tests here.
    {
        const int w = tid;
        float s[5];
        #pragma unroll
        for (int c = 0; c < 5; ++c) {
            float a = 0.f;
            #pragma unroll
            for (int r = 0; r < 9; ++r) a += chan[c][r][w];
            s[c] = a;
        }
        const size_t obase = (((size_t)b * D_ + d) * H_ + h0) * W_ + w;
        #pragma unroll
        for (int ro = 0; ro < TH; ++ro) {
            if (ro > 0) {
                #pragma unroll
                for (int c = 0; c < 5; ++c)
                    s[c] += chan[c][ro + 8][w] - chan[c][ro - 1][w];
            }
            #pragma unroll
            for (int c = 0; c < 5; ++c)
                ws[(size_t)c * NVOX + obase + (size_t)ro * W_] = s[c];
        }
    }
}

// =================================================================
// Kernel 2: D-direction running box sum of the 5 channels + cc + partial sum.
// One thread marches a 48-deep segment of one (b,h,w) column.
// The re-read at d-4 is only 9 planes (~1.3 MB/channel) behind the streaming
// front: L2-resident (192 MB L2), so HBM traffic stays ~1 read per element.
// =================================================================
__device__ __forceinline__ float lcc_cc(const float S[5]) {
    const float inv_win = 1.0f / 729.0f;
    const float Isum = S[0], Jsum = S[1];
    const float uI = Isum * inv_win, uJ = Jsum * inv_win;
    const float cross = S[4] - uJ * Isum - uI * Jsum + uI * uJ * 729.0f;
    const float Ivar  = S[2] - 2.0f * uI * Isum + uI * uI * 729.0f;
    const float Jvar  = S[3] - 2.0f * uJ * Jsum + uJ * uJ * 729.0f;
    return (cross * cross) / (Ivar * Jvar + 1e-5f);
}

__global__ __launch_bounds__(K2_THREADS) void lcc_d_kernel(
        const float* __restrict__ ws, float* __restrict__ partials)
{
    const int t   = blockIdx.x * K2_THREADS + threadIdx.x;
    const int w   = t % W_;
    const int h   = (t / W_) % H_;
    const int seg = (t / (H_ * W_)) % NSEG;
    const int b   = t / (NSEG * H_ * W_);
    const int d0  = seg * SEG;

    const size_t base = (size_t)b * D_ * PL + (size_t)h * W_ + w;
    const float* ch[5];
    #pragma unroll
    for (int c = 0; c < 5; ++c) ch[c] = ws + (size_t)c * NVOX;

    float S[5] = {0.f, 0.f, 0.f, 0.f, 0.f};
    #pragma unroll
    for (int k = -4; k <= 4; ++k) {
        const int kk = d0 + k;
        if (kk >= 0 && kk < D_) {
            #pragma unroll
            for (int c = 0; c < 5; ++c) S[c] += ch[c][base + (size_t)kk * PL];
        }
    }

    float acc = 0.f;
    // Edge predicates are wave-uniform in d; split the loop so the steady
    // state does unconditional add/sub. kr = d-4 < 0 only for d <= 3 (first
    // segment); ka = d+5 > 191 only for d >= 187 (last segment).
    const int d_lo = (d0 == 0) ? 4 : d0;                 // first d with kr>=0
    const int d_hi = (d0 + SEG == D_) ? (D_ - 5) : (d0 + SEG);  // first d with ka oob

    for (int d = d0; d < d_lo; ++d) {                    // prologue (seg 0 only)
        acc += lcc_cc(S);
        #pragma unroll
        for (int c = 0; c < 5; ++c) S[c] += ch[c][base + (size_t)(d + 5) * PL];
    }
    for (int d = d_lo; d < d_hi; ++d) {                  // steady state
        acc += lcc_cc(S);
        const size_t pa = base + (size_t)(d + 5) * PL;
        const size_t pr = base + (size_t)(d - 4) * PL;
        // Pull the stream front into cache ahead of use (global_prefetch_b8).
        if (d + 5 + PFD < D_) {
            #pragma unroll
            for (int c = 0; c < 5; ++c)
                __builtin_prefetch(&ch[c][pa + (size_t)PFD * PL], 0, 1);
        }
        #pragma unroll
        for (int c = 0; c < 5; ++c) S[c] += ch[c][pa];
        #pragma unroll
        for (int c = 0; c < 5; ++c) S[c] -= ch[c][pr];
    }
    for (int d = d_hi; d < d0 + SEG; ++d) {              // epilogue (last seg only)
        acc += lcc_cc(S);
        #pragma unroll
        for (int c = 0; c < 5; ++c) S[c] -= ch[c][base + (size_t)(d - 4) * PL];
    }

    __shared__ float red[K2_THREADS];
    red[threadIdx.x] = acc;
    __syncthreads();
    #pragma unroll
    for (int s = K2_THREADS / 2; s > 0; s >>= 1) {
        if (threadIdx.x < s) red[threadIdx.x] += red[threadIdx.x + s];
        __syncthreads();
    }
    if (threadIdx.x == 0) partials[blockIdx.x] = red[0];
}

// =================================================================
// Kernel 3: deterministic final reduction (no float atomics -> bit-stable
// across graph replays). out = -mean(cc) + 1.
// =================================================================
__global__ __launch_bounds__(256) void lcc_final_kernel(
        const float* __restrict__ partials, float* __restrict__ out)
{
    __shared__ float red[256];
    float a = 0.f;
    for (int i = threadIdx.x; i < K2_BLOCKS; i += 256) a += partials[i];
    red[threadIdx.x] = a;
    __syncthreads();
    #pragma unroll
    for (int s = 128; s > 0; s >>= 1) {
        if (threadIdx.x < s) red[threadIdx.x] += red[threadIdx.x + s];
        __syncthreads();
    }
    if (threadIdx.x == 0)
        out[0] = 1.0f - red[0] * (1.0f / (float)NVOX);
}

// =================================================================
// Host entry. Workspace layout:
//   ws[0 .. 5*NVOX)            : 5 intermediate channel volumes (f32), ~283 MB
//   ws[5*NVOX .. +K2_BLOCKS)   : per-block partial sums
// =================================================================
extern "C" void kernel_launch(void* const* d_in, const int* in_sizes, int n_in,
                              void* d_out, int out_size, void* d_ws, size_t ws_size,
                              hipStream_t stream)
{
    (void)in_sizes; (void)n_in; (void)out_size; (void)ws_size;
    const float* I = (const float*)d_in[0];
    const float* J = (const float*)d_in[1];
    float* ws = (float*)d_ws;
    float* partials = ws + 5 * NVOX;

    lcc_wh_kernel<<<K1_BLOCKS, K1_THREADS, 0, stream>>>(I, J, ws);
    lcc_d_kernel<<<K2_BLOCKS, K2_THREADS, 0, stream>>>(ws, partials);
    lcc_final_kernel<<<1, 256, 0, stream>>>(partials, (float*)d_out);
}